// MyGATv2Encoder_17738214932692
// MI455X (gfx1250) — compile-verified
//
#include <hip/hip_runtime.h>
#include <math.h>

#define NNODES 50000
#define E_REAL 800000
#define E_TOT  850000

typedef float v2f __attribute__((ext_vector_type(2)));
typedef float v8f __attribute__((ext_vector_type(8)));

#if defined(__HIP_DEVICE_COMPILE__)
#if __has_builtin(__builtin_amdgcn_wmma_f32_16x16x4_f32)
#define USE_WMMA_F32 1
#else
#define USE_WMMA_F32 0
#endif
#else
#define USE_WMMA_F32 0
#endif

__device__ __forceinline__ int imin(int a, int b) { return a < b ? a : b; }

// ---------------------------------------------------------------------------
// GEMM: Y[M x F] = X[M x K] @ W[K x F] + bias[F]   (row-major everywhere)
// One wave computes a 32x64 output tile: 2 row-tiles (A fragments) x 4
// col-tiles (B fragments), 8 f32-WMMA 16x16x4 per k-step. Bounds handling is
// hoisted out of the K loop (addresses clamped; OOB columns accumulate
// garbage that is never stored) and the loop is software-pipelined: the 10
// loads for step k+4 are issued before the 8 WMMAs consuming step k, so the
// matrix ops hide the L2 load latency. K%4 tail handled in an epilogue.
// ---------------------------------------------------------------------------
__global__ __launch_bounds__(256)
void k_gemm_wmma(const float* __restrict__ X, const float* __restrict__ W,
                 const float* __restrict__ bias, float* __restrict__ Y,
                 int M, int K, int F) {
  const int lane    = threadIdx.x & 31;
  const int wave    = threadIdx.x >> 5;
  const int pair    = blockIdx.y * 8 + wave;   // 32-row tile index
  const int m0      = pair * 32;
  if (m0 >= M) return;
  const int n0      = blockIdx.x * 64;
  const int halfSel = lane >> 4;               // 0: lanes 0-15, 1: lanes 16-31
  const int ln      = lane & 15;
  const int kOff    = halfSel * 2;
  const bool tile1ok = (m0 + 16) < M;

  // A fragment rows (clamped for tile 1; its store is guarded)
  const int aRow0 = m0 + ln;
  const int aRow1 = imin(m0 + 16 + ln, M - 1);

  // Column addresses, clamped for loads; store predicate kept separately
  int  colc[4];
  bool cok[4];
#pragma unroll
  for (int t = 0; t < 4; ++t) {
    const int col = n0 + t * 16 + ln;
    cok[t]  = (col < F);
    colc[t] = imin(col, F - 1);
  }

  v8f acc0[4], acc1[4];
#pragma unroll
  for (int t = 0; t < 4; ++t) {
    acc0[t] = (v8f){0.f,0.f,0.f,0.f,0.f,0.f,0.f,0.f};
    acc1[t] = (v8f){0.f,0.f,0.f,0.f,0.f,0.f,0.f,0.f};
  }

  const int K4 = K & ~3;

#if USE_WMMA_F32
  // A fragment (16x4 f32): lanes 0-15 hold K=k..k+1, lanes 16-31 K=k+2..k+3
  // B fragment (4x16 f32): VGPR0 -> K=ka, VGPR1 -> K=ka+1, N = lane&15
  auto loadFrags = [&](int kbase, v2f& a0f, v2f& a1f, v2f* bf) {
    const int ka = kbase + kOff;
    a0f = *(const v2f*)(X + aRow0 * K + ka);
    a1f = *(const v2f*)(X + aRow1 * K + ka);
#pragma unroll
    for (int t = 0; t < 4; ++t) {
      const int idx = ka * F + colc[t];
      bf[t].x = W[idx];
      bf[t].y = W[idx + F];
    }
  };
  auto doWmma = [&](const v2f& a0f, const v2f& a1f, const v2f* bf) {
#pragma unroll
    for (int t = 0; t < 4; ++t) {
      acc0[t] = __builtin_amdgcn_wmma_f32_16x16x4_f32(
          false, a0f, false, bf[t], (short)0, acc0[t], false, false);
      acc1[t] = __builtin_amdgcn_wmma_f32_16x16x4_f32(
          false, a1f, false, bf[t], (short)0, acc1[t], false, false);
    }
  };

  v2f a0c, a1c, bc[4];
  loadFrags(0, a0c, a1c, bc);
#pragma unroll 2
  for (int k0 = 0; k0 < K4; k0 += 4) {
    // Prefetch next step (clamped: last iteration harmlessly re-reads k=0)
    const int kn = (k0 + 4 < K4) ? (k0 + 4) : 0;
    v2f a0n, a1n, bn[4];
    loadFrags(kn, a0n, a1n, bn);
    // Consume current registers while prefetch is in flight
    doWmma(a0c, a1c, bc);
    a0c = a0n;
    a1c = a1n;
#pragma unroll
    for (int t = 0; t < 4; ++t) bc[t] = bn[t];
  }

  if (K4 < K) {  // K % 4 == 2 for K = 306, 198
    const int ka  = K4 + kOff;
    const bool k0ok = (ka < K), k1ok = (ka + 1 < K);
    v2f a0, a1;
    a0.x = k0ok ? X[aRow0 * K + ka]     : 0.f;
    a0.y = k1ok ? X[aRow0 * K + ka + 1] : 0.f;
    a1.x = k0ok ? X[aRow1 * K + ka]     : 0.f;
    a1.y = k1ok ? X[aRow1 * K + ka + 1] : 0.f;
    v2f bt[4];
#pragma unroll
    for (int t = 0; t < 4; ++t) {
      const int idx = ka * F + colc[t];
      bt[t].x = k0ok ? W[idx]     : 0.f;
      bt[t].y = k1ok ? W[idx + F] : 0.f;
    }
    doWmma(a0, a1, bt);
  }
#else
  // Scalar fallback with identical per-lane fragment ownership
#pragma unroll
  for (int t = 0; t < 4; ++t) {
    for (int r = 0; r < 8; ++r) {
      const int row0 = m0 + halfSel * 8 + r;
      const int row1 = imin(m0 + 16 + halfSel * 8 + r, M - 1);
      float s0 = 0.f, s1 = 0.f;
      for (int k = 0; k < K; ++k) {
        const float w = W[k * F + colc[t]];
        s0 += X[row0 * K + k] * w;
        s1 += X[row1 * K + k] * w;
      }
      acc0[t][r] = s0;
      acc1[t][r] = s1;
    }
  }
#endif

  // C/D layout: VGPR r -> M = base + r + 8*halfSel, N = lane&15
#pragma unroll
  for (int t = 0; t < 4; ++t) {
    if (!cok[t]) continue;
    const int col = n0 + t * 16 + ln;
    const float bs = bias[col];
    const int rb0 = m0 + halfSel * 8;
#pragma unroll
    for (int r = 0; r < 8; ++r)
      Y[(rb0 + r) * F + col] = acc0[t][r] + bs;
    if (tile1ok) {
      const int rb1 = m0 + 16 + halfSel * 8;
#pragma unroll
      for (int r = 0; r < 8; ++r)
        Y[(rb1 + r) * F + col] = acc1[t][r] + bs;
    }
  }
}

// ---------------------------------------------------------------------------
// Utility fills
// ---------------------------------------------------------------------------
__global__ void k_fill_f32(float* __restrict__ p, int n, float v) {
  int t = blockIdx.x * blockDim.x + threadIdx.x;
  if (t < n) p[t] = v;
}
__global__ void k_fill_u32(unsigned* __restrict__ p, int n, unsigned v) {
  int t = blockIdx.x * blockDim.x + threadIdx.x;
  if (t < n) p[t] = v;
}

// Ordered-uint encoding for float atomicMax (total order, handles negatives)
__device__ __forceinline__ unsigned f2ord(float f) {
  unsigned u = __float_as_uint(f);
  return (u & 0x80000000u) ? ~u : (u | 0x80000000u);
}
__device__ __forceinline__ float ord2f(unsigned u) {
  unsigned v = (u & 0x80000000u) ? (u ^ 0x80000000u) : ~u;
  return __uint_as_float(v);
}

__device__ __forceinline__ void edge_src_dst(const long long* __restrict__ ei,
                                             int e, int& src, int& dst) {
  if (e < E_REAL) {
    src = (int)ei[e];
    dst = (int)ei[E_REAL + e];
  } else {
    src = dst = e - E_REAL;   // appended self-loops
  }
}

// ---------------------------------------------------------------------------
// Pass 1: per (edge, head) attention logit + segment max (atomic, ordered-uint)
// logits[e,h] = att[h,:] . leaky_relu(xl[src,h,:] + xr[dst,h,:], 0.2)
// ---------------------------------------------------------------------------
__global__ void k_edge_logits(const float* __restrict__ Q, const float* __restrict__ R,
                              const float* __restrict__ att, const long long* __restrict__ ei,
                              float* __restrict__ logits, unsigned* __restrict__ menc,
                              int H, int C, int F) {
  int t = blockIdx.x * blockDim.x + threadIdx.x;
  if (t >= E_TOT * H) return;
  int e = t / H, h = t - e * H;
  int src, dst;
  edge_src_dst(ei, e, src, dst);
  const float* q = Q + src * F + h * C;
  const float* r = R + dst * F + h * C;
  const float* a = att + h * C;
  float acc = 0.f;
  for (int c = 0; c < C; ++c) {
    float v = q[c] + r[c];
    v = (v > 0.f) ? v : 0.2f * v;
    acc += a[c] * v;
  }
  logits[t] = acc;
  atomicMax(&menc[dst * H + h], f2ord(acc));
}

// ---------------------------------------------------------------------------
// Pass 2: ex = exp(logit - max[dst]); denom[dst] += ex (segment sum)
// ---------------------------------------------------------------------------
__global__ void k_edge_exp(float* __restrict__ logits, const unsigned* __restrict__ menc,
                           float* __restrict__ den, const long long* __restrict__ ei,
                           int H) {
  int t = blockIdx.x * blockDim.x + threadIdx.x;
  if (t >= E_TOT * H) return;
  int e = t / H, h = t - e * H;
  int src, dst;
  edge_src_dst(ei, e, src, dst);
  (void)src;
  float ex = expf(logits[t] - ord2f(menc[dst * H + h]));
  logits[t] = ex;
  atomicAdd(&den[dst * H + h], ex);
}

// ---------------------------------------------------------------------------
// Pass 3: out[dst,h,c] += (ex/denom[dst,h]) * xl[src,h,c]
// One block per edge; F-wide thread block; alpha cached in LDS per head.
// Gathers/atomics land in L2 (activation matrix fits in 192 MB L2).
// ---------------------------------------------------------------------------
__global__ void k_edge_scatter(const float* __restrict__ Q, const float* __restrict__ ex,
                               const float* __restrict__ den, const long long* __restrict__ ei,
                               float* __restrict__ P, int H, int C, int F) {
  __shared__ float alpha[8];
  const int e = blockIdx.x;
  const int f = threadIdx.x;
  int src, dst;
  edge_src_dst(ei, e, src, dst);
  if (f < H) alpha[f] = ex[e * H + f] / den[dst * H + f];
  __syncthreads();
  if (f < F) {
    int h = f / C;
    atomicAdd(&P[dst * F + f], alpha[h] * Q[src * F + f]);
  }
}

// ---------------------------------------------------------------------------
// Finalize (concat layers): P = elu(batchnorm(P + b_conv))
// ---------------------------------------------------------------------------
__global__ void k_finalize_concat(float* __restrict__ P, const float* __restrict__ bconv,
                                  const float* __restrict__ g, const float* __restrict__ bb,
                                  const float* __restrict__ rm, const float* __restrict__ rv,
                                  int F, int total) {
  int t = blockIdx.x * blockDim.x + threadIdx.x;
  if (t >= total) return;
  int f = t % F;
  float scale = g[f] * rsqrtf(rv[f] + 1e-5f);
  float v = (P[t] + bconv[f] - rm[f]) * scale + bb[f];
  P[t] = (v > 0.f) ? v : expm1f(v);   // ELU, alpha = 1
}

// ---------------------------------------------------------------------------
// Finalize (last layer, concat=False): mean over 5 heads, + b, BN -> d_out
// ---------------------------------------------------------------------------
__global__ void k_finalize_mean(const float* __restrict__ P, const float* __restrict__ bconv,
                                const float* __restrict__ g, const float* __restrict__ bb,
                                const float* __restrict__ rm, const float* __restrict__ rv,
                                float* __restrict__ out) {
  int t = blockIdx.x * blockDim.x + threadIdx.x;
  if (t >= NNODES * 64) return;
  int n = t >> 6, c = t & 63;
  const float* row = P + n * 320;
  float s = 0.f;
#pragma unroll
  for (int h = 0; h < 5; ++h) s += row[h * 64 + c];
  s = s * 0.2f + bconv[c];
  out[t] = (s - rm[c]) * (g[c] * rsqrtf(rv[c] + 1e-5f)) + bb[c];
}

// ---------------------------------------------------------------------------
static inline int ceil_div(int a, int b) { return (a + b - 1) / b; }

extern "C" void kernel_launch(void* const* d_in, const int* in_sizes, int n_in,
                              void* d_out, int out_size, void* d_ws, size_t ws_size,
                              hipStream_t stream) {
  (void)in_sizes; (void)n_in; (void)out_size; (void)ws_size;

  const float*     x  = (const float*)d_in[0];
  const long long* ei = (const long long*)d_in[1];   // edge_index is int64 [2, 800000]

  // Workspace layout (floats):
  float*    Q  = (float*)d_ws;                       // xl  : N x 320 (max F)
  float*    R  = Q  + (size_t)NNODES * 320;          // xr  : N x 320
  float*    P  = R  + (size_t)NNODES * 320;          // accum / activations : N x 320
  float*    EX = P  + (size_t)NNODES * 320;          // per-edge logits/ex : E_TOT x 5
  unsigned* ME = (unsigned*)(EX + (size_t)E_TOT * 5);// segment max (encoded) : N x 5
  float*    DE = (float*)(ME + (size_t)NNODES * 5);  // segment denom : N x 5

  struct LayerCfg { int K, H, C, ic, ib; };
  const LayerCfg L[4] = {
      {128, 3, 102,  2,  8},   // conv1 / bn1
      {306, 3,  76, 12, 18},   // conv2 / bn2
      {228, 3,  66, 22, 28},   // conv3 / bn3
      {198, 5,  64, 32, 38},   // conv4 / bn4 (mean over heads)
  };

  const float* in = x;
  for (int l = 0; l < 4; ++l) {
    const int K = L[l].K, H = L[l].H, C = L[l].C, F = H * C;
    const float* Wl  = (const float*)d_in[L[l].ic + 0];
    const float* bl  = (const float*)d_in[L[l].ic + 1];
    const float* Wr  = (const float*)d_in[L[l].ic + 2];
    const float* br  = (const float*)d_in[L[l].ic + 3];
    const float* att = (const float*)d_in[L[l].ic + 4];
    const float* bc  = (const float*)d_in[L[l].ic + 5];
    const float* g   = (const float*)d_in[L[l].ib + 0];
    const float* bb  = (const float*)d_in[L[l].ib + 1];
    const float* rm  = (const float*)d_in[L[l].ib + 2];
    const float* rv  = (const float*)d_in[L[l].ib + 3];

    // Dense projections (WMMA f32): xl = in@Wl + bl ; xr = in@Wr + br
    dim3 gg(ceil_div(F, 64), ceil_div(ceil_div(NNODES, 32), 8));
    k_gemm_wmma<<<gg, 256, 0, stream>>>(in, Wl, bl, Q, NNODES, K, F);
    k_gemm_wmma<<<gg, 256, 0, stream>>>(in, Wr, br, R, NNODES, K, F);

    // Reset segment reductions + output accumulator (gemm above already
    // consumed P as input; stream order makes this safe).
    const int nh = NNODES * H;
    k_fill_u32<<<ceil_div(nh, 256), 256, 0, stream>>>(ME, nh, 0u);
    k_fill_f32<<<ceil_div(nh, 256), 256, 0, stream>>>(DE, nh, 0.f);
    k_fill_f32<<<ceil_div(NNODES * F, 256), 256, 0, stream>>>(P, NNODES * F, 0.f);

    // Edge-wise softmax attention
    const int eth = E_TOT * H;
    k_edge_logits<<<ceil_div(eth, 256), 256, 0, stream>>>(Q, R, att, ei, EX, ME, H, C, F);
    k_edge_exp   <<<ceil_div(eth, 256), 256, 0, stream>>>(EX, ME, DE, ei, H);
    const int bs = ((F + 63) / 64) * 64;
    k_edge_scatter<<<E_TOT, bs, 0, stream>>>(Q, EX, DE, ei, P, H, C, F);

    if (l < 3) {
      k_finalize_concat<<<ceil_div(NNODES * F, 256), 256, 0, stream>>>(
          P, bc, g, bb, rm, rv, F, NNODES * F);
      in = P;  // next layer consumes the activated output
    } else {
      k_finalize_mean<<<ceil_div(NNODES * 64, 256), 256, 0, stream>>>(
          P, bc, g, bb, rm, rv, (float*)d_out);
    }
  }
}